// TPe3nn_41197326303991
// MI455X (gfx1250) — compile-verified
//
#include <hip/hip_runtime.h>
#include <hip/hip_bf16.h>

typedef __attribute__((ext_vector_type(2))) float v2f;
typedef __attribute__((ext_vector_type(4))) float v4f;
typedef __attribute__((ext_vector_type(8))) float v8f;

#define KDIM   4096          // u*v flattened GEMM K
#define OUTLD  65536
#define NCHUNK (KDIM / 32)   // 128 K-chunks of 32

// alpha * C collapses to two constants everywhere:
//   1/sqrt(8192)  = all "c" couplings
//   1/sqrt(24576) = trace-type couplings (delta/sqrt3 on 0e, z^2 row of W112)
#define C_A 0.01104854345603981f   // 1/sqrt(8192)
#define C_B 0.00637887357223024f   // 1/sqrt(24576)

// T workspace offsets (floats): rows = B*KP, cols = 4096, row-major
#define T1_OFF 0u          // 128*4096
#define T2_OFF 524288u
#define T3_OFF 1048576u    // 384*4096
#define T4_OFF 2621440u
#define T5_OFF 4194304u    // 384*4096
#define T6_OFF 5767168u    // 640*4096

// ---------------------------------------------------------------------------
// Kernel 1: build T_p[(b*KP+k), u*64+v] = alpha_p * sum_ij xa[b,u,i] xb[b,v,j] C_p[i,j,k]
// ---------------------------------------------------------------------------
__global__ __launch_bounds__(256) void build_t_kernel(
    const float* __restrict__ x1, const float* __restrict__ x2,
    float* __restrict__ ws)
{
    const int path = blockIdx.y;                 // 0..5
    const long gid = (long)blockIdx.x * 256 + threadIdx.x;

    int Mp, KP; unsigned off;
    switch (path) {
        case 0: Mp = 128; KP = 1; off = T1_OFF; break;
        case 1: Mp = 128; KP = 1; off = T2_OFF; break;
        case 2: Mp = 384; KP = 3; off = T3_OFF; break;
        case 3: Mp = 384; KP = 3; off = T4_OFF; break;
        case 4: Mp = 384; KP = 3; off = T5_OFF; break;
        default:Mp = 640; KP = 5; off = T6_OFF; break;
    }
    if (gid >= (long)Mp * KDIM) return;

    const int row = (int)(gid >> 12);            // (b,k) row
    const int uv  = (int)(gid & 4095);
    const int u   = uv >> 6;
    const int v   = uv & 63;
    const int b   = row / KP;
    const int k   = row - b * KP;

    const float* __restrict__ xa = x1 + (size_t)b * 256;
    const float* __restrict__ xb = x2 + (size_t)b * 256;

    float val;
    if (path == 0) {                                   // (0e,0e)->0e
        val = C_A * xa[u] * xb[v];
    } else if (path == 1) {                            // (1o,1o)->0e : delta_ij/sqrt3
        const float* a = xa + 64 + u * 3;
        const float* c = xb + 64 + v * 3;
        val = C_B * (a[0]*c[0] + a[1]*c[1] + a[2]*c[2]);
    } else if (path == 2) {                            // (0e,1o)->1o
        val = C_A * xa[u] * xb[64 + v * 3 + k];
    } else if (path == 3) {                            // (1o,0e)->1o
        val = C_A * xa[64 + u * 3 + k] * xb[v];
    } else if (path == 4) {                            // (1o,1o)->1e : Levi-Civita/sqrt6
        const float* a = xa + 64 + u * 3;
        const float* c = xb + 64 + v * 3;
        float cr;
        if      (k == 0) cr = a[1]*c[2] - a[2]*c[1];
        else if (k == 1) cr = a[2]*c[0] - a[0]*c[2];
        else             cr = a[0]*c[1] - a[1]*c[0];
        val = C_A * cr;
    } else {                                           // (1o,1o)->2e : quadrupole coupling
        const float* a = xa + 64 + u * 3;              // basis order (y,z,x)
        const float* c = xb + 64 + v * 3;
        if      (k == 0) val = C_A * (a[2]*c[0] + a[0]*c[2]);                 // xy
        else if (k == 1) val = C_A * (a[0]*c[1] + a[1]*c[0]);                 // yz
        else if (k == 2) val = C_B * (2.f*a[1]*c[1] - a[0]*c[0] - a[2]*c[2]); // z^2
        else if (k == 3) val = C_A * (a[1]*c[2] + a[2]*c[1]);                 // xz
        else             val = C_A * (a[2]*c[2] - a[0]*c[0]);                 // x^2-y^2
    }
    ws[off + (size_t)gid] = val;
}

// ---------------------------------------------------------------------------
// Kernel 2: D = T0@W0 (+ T1@W1), D[(b,k), n] -> out[b*65536 + base + n*KP + k]
// 256 threads = 8 waves: 2 N-columns(16) x 4 M-quarters.  Full-M per workgroup
// so each weight byte crosses HBM exactly once; weight tiles stream through a
// double-buffered LDS stage with non-temporal loads (T stays L2-resident).
// WMMA f32 16x16x4, K-chunk = 32 (8 wmma k-steps per chunk), 1 barrier/chunk.
// ---------------------------------------------------------------------------
template<int MT, int KP>   // MT = M-tiles(16) per wave; total M = MT*64
__global__ __launch_bounds__(256) void tp_gemm_kernel(
    const float* __restrict__ T0, const float* __restrict__ W0,
    const float* __restrict__ T1, const float* __restrict__ W1,
    int Nw, float* __restrict__ out, int outBase)
{
    __shared__ __align__(16) float lds_w[2][32][36];   // ping-pong 32x32 W tiles

    const int tid  = threadIdx.x;
    const int wave = tid >> 5;
    const int lane = tid & 31;
    const int h    = lane >> 4;          // half-wave: K pair select
    const int l16  = lane & 15;
    const int ncol = (wave & 1) * 16;    // column-16 within 32-wide tile
    const int mq   = wave >> 1;          // M quarter 0..3
    const int mbase = mq * (MT * 16);
    const int n0   = blockIdx.x * 32;

    // cooperative W-load role: 32 rows x (8 lanes * float4)
    const int wr = tid >> 3;
    const int wc = (tid & 7) * 4;

    v8f acc[MT];
#pragma unroll
    for (int mt = 0; mt < MT; ++mt) acc[mt] = (v8f)0.f;

    for (int p = 0; p < 2; ++p) {
        const float* __restrict__ Tc = (p == 0) ? T0 : T1;
        const float* __restrict__ Wc = (p == 0) ? W0 : W1;
        if (Tc == nullptr) break;        // uniform across block

        const float* wsrc = Wc + (size_t)wr * Nw + n0 + wc;

        // prologue: stage chunk 0
        {
            const v4f w0v = __builtin_nontemporal_load((const v4f*)wsrc);
            *(v4f*)&lds_w[0][wr][wc] = w0v;
        }
        __syncthreads();

        for (int ci = 0; ci < NCHUNK; ++ci) {
            const int k0  = ci * 32;
            const int cur = ci & 1;

            // prefetch next W chunk (overlaps with WMMA compute below)
            v4f wnext;
            const bool have_next = (ci + 1 < NCHUNK);
            if (have_next)
                wnext = __builtin_nontemporal_load(
                    (const v4f*)(wsrc + (size_t)(k0 + 32) * Nw));

#pragma unroll
            for (int ks = 0; ks < 8; ++ks) {
                const int kk = ks * 4;
                // B fragment (4x16): vgpr q holds row K = 2h+q, N = ncol+l16
                v2f bf;
                bf.x = lds_w[cur][kk + 2 * h][ncol + l16];
                bf.y = lds_w[cur][kk + 2 * h + 1][ncol + l16];
#pragma unroll
                for (int mt = 0; mt < MT; ++mt) {
                    // A fragment (16x4): lane holds M = l16, K = {2h, 2h+1}
                    const int row = mbase + mt * 16 + l16;
                    const v2f af =
                        *(const v2f*)(Tc + (size_t)row * KDIM + k0 + kk + 2 * h);
                    acc[mt] = __builtin_amdgcn_wmma_f32_16x16x4_f32(
                        false, af, false, bf, (short)0, acc[mt], false, false);
                }
            }

            if (have_next)
                *(v4f*)&lds_w[1 - cur][wr][wc] = wnext;
            __syncthreads();   // stores of next buffer visible; readers of cur done
        }
    }

    // C/D layout: vgpr r, half h -> M = r + 8h ; N = ncol + l16
    const int ng = n0 + ncol + l16;
#pragma unroll
    for (int mt = 0; mt < MT; ++mt) {
#pragma unroll
        for (int r = 0; r < 8; ++r) {
            const int m = mbase + mt * 16 + 8 * h + r;   // (b,k) row
            const int b = m / KP;
            const int k = m - b * KP;
            __builtin_nontemporal_store(
                acc[mt][r],
                &out[(size_t)b * OUTLD + outBase + (size_t)ng * KP + k]);
        }
    }
}

// ---------------------------------------------------------------------------
extern "C" void kernel_launch(void* const* d_in, const int* in_sizes, int n_in,
                              void* d_out, int out_size, void* d_ws, size_t ws_size,
                              hipStream_t stream) {
    const float* x1 = (const float*)d_in[0];
    const float* x2 = (const float*)d_in[1];
    const float* w1 = (const float*)d_in[2];
    const float* w2 = (const float*)d_in[3];
    const float* w3 = (const float*)d_in[4];
    const float* w4 = (const float*)d_in[5];
    const float* w5 = (const float*)d_in[6];
    const float* w6 = (const float*)d_in[7];
    float* out = (float*)d_out;
    float* ws  = (float*)d_ws;           // needs 32 MiB for T tensors

    // Build all T tensors (biggest path: 640*4096 elems -> 10240 blocks)
    build_t_kernel<<<dim3(10240, 6), 256, 0, stream>>>(x1, x2, ws);

    // o0  = T1@W1 + T2@W2 : M=128 (MT=2), N=8192, KP=1, base 0
    tp_gemm_kernel<2, 1><<<8192 / 32, 256, 0, stream>>>(
        ws + T1_OFF, w1, ws + T2_OFF, w2, 8192, out, 0);
    // o1  = T3@W3 + T4@W4 : M=384 (MT=6), N=8192, KP=3, base 8192
    tp_gemm_kernel<6, 3><<<8192 / 32, 256, 0, stream>>>(
        ws + T3_OFF, w3, ws + T4_OFF, w4, 8192, out, 8192);
    // o1e = T5@W5 : M=384 (MT=6), N=4096, KP=3, base 32768
    tp_gemm_kernel<6, 3><<<4096 / 32, 256, 0, stream>>>(
        ws + T5_OFF, w5, nullptr, nullptr, 4096, out, 32768);
    // o2e = T6@W6 : M=640 (MT=10), N=4096, KP=5, base 45056
    tp_gemm_kernel<10, 5><<<4096 / 32, 256, 0, stream>>>(
        ws + T6_OFF, w6, nullptr, nullptr, 4096, out, 45056);
}